// CubPL2d_7310034338282
// MI455X (gfx1250) — compile-verified
//
#include <hip/hip_runtime.h>
#include <stdint.h>

// CubPL2d persistence-landscape kernel for MI455X (gfx1250, wave32).
//
// Strategy (memory-latency-bound problem, ~24MB useful traffic):
//  - 1 workgroup (128 thr = 4 waves) per (b,c) channel; 8192 blocks.
//  - Phase 1: data-dependent gathers of births/deaths go straight to LDS via
//    CDNA5 async global->LDS (ASYNCcnt path), no VGPR round-trip.
//  - Phase 2: tri[p][t] = max(min(t-birth, death-t), 0) computed as outer
//    sums with V_WMMA_F32_16X16X4_F32 (A = value bcast over K, B = +/-0.25
//    everywhere so sum_k A*B = +/-A exactly, C = +/-t bcast per column).
//    16 (p,t) tiles of 16x16 cover the 128x32 surface; fp32-exact.
//  - Phase 3: 64 threads (one per (d,t)) do the dim-masked top-2 over p
//    from LDS (broadcast dim reads, stride-1 tri reads -> conflict-free).

#define BC_TOTAL  (128 * 64)
#define HW_       4096
#define P_        128
#define T_        32
#define T_MIN_    0.03f
#define T_STEP_   (0.31f / 31.0f)   // (0.34-0.03)/(32-1)

typedef __attribute__((ext_vector_type(2))) float v2f;
typedef __attribute__((ext_vector_type(8))) float v8f;

__global__ __launch_bounds__(128) void CubPL2d_kernel(
    const float* __restrict__ x,
    const int*   __restrict__ birth_idx,
    const int*   __restrict__ death_idx,
    const int*   __restrict__ pair_dim,
    float*       __restrict__ out)
{
    __shared__ float birth_s[P_];
    __shared__ float death_s[P_];
    __shared__ int   dim_s[P_];
    __shared__ float tri_s[P_ * T_];   // [p][t], 16 KB

    const int bc  = blockIdx.x;
    const int tid = threadIdx.x;

    // ---------------- Phase 1: async gathers into LDS ----------------
    {
        const int p  = tid;                     // 128 threads == P_
        const size_t ib = (size_t)bc * P_ + p;
        const int bi = birth_idx[ib];
        const int di = death_idx[ib];
        dim_s[p] = pair_dim[ib];

        const float* xb = x + (size_t)bc * HW_;
        unsigned long long ga_b = (unsigned long long)(uintptr_t)(xb + bi);
        unsigned long long ga_d = (unsigned long long)(uintptr_t)(xb + di);
        // Flat LDS addresses keep the byte offset in the low 32 bits.
        unsigned lds_b = (unsigned)(uintptr_t)&birth_s[p];
        unsigned lds_d = (unsigned)(uintptr_t)&death_s[p];

        asm volatile("global_load_async_to_lds_b32 %0, %1, off"
                     :: "v"(lds_b), "v"(ga_b) : "memory");
        asm volatile("global_load_async_to_lds_b32 %0, %1, off"
                     :: "v"(lds_d), "v"(ga_d) : "memory");
        asm volatile("s_wait_asynccnt 0" ::: "memory");
    }
    __syncthreads();

    // ------- Phase 2: tri surface via V_WMMA_F32_16X16X4_F32 tiles -------
    {
        const int wave = tid >> 5;
        const int lane = tid & 31;
        const int mrow = lane & 15;   // M index (and N index) within a tile
        const int half = lane >> 4;   // D-matrix: rows M = r + 8*half

        v2f bneg; bneg.x = -0.25f; bneg.y = -0.25f;  // sum_k a*(-1/4) = -a (exact)
        v2f bpos; bpos.x =  0.25f; bpos.y =  0.25f;  // sum_k a*(+1/4) = +a (exact)

#pragma unroll
        for (int i = 0; i < 2; ++i) {
            const int p0 = (wave * 2 + i) * 16;      // waves cover p-tiles 0..7
            const float bv = birth_s[p0 + mrow];     // A[m,k] = birth[p0+m], all k
            const float dv = death_s[p0 + mrow];
            v2f a1; a1.x = bv; a1.y = bv;
            v2f a2; a2.x = dv; a2.y = dv;
#pragma unroll
            for (int j = 0; j < 2; ++j) {
                const int t0 = j * 16;
                const float tv = T_MIN_ + (float)(t0 + mrow) * T_STEP_;
                v8f c1, c2;
#pragma unroll
                for (int r = 0; r < 8; ++r) { c1[r] = tv; c2[r] = -tv; }

                // d1[m][n] = t[n] - birth[m] ; d2[m][n] = death[m] - t[n]
                v8f d1 = __builtin_amdgcn_wmma_f32_16x16x4_f32(
                    false, a1, false, bneg, (short)0, c1, false, false);
                v8f d2 = __builtin_amdgcn_wmma_f32_16x16x4_f32(
                    false, a2, false, bpos, (short)0, c2, false, false);

                const int prow = p0 + half * 8;      // D layout: M = r + 8*half
                const int tcol = t0 + mrow;          //           N = lane & 15
#pragma unroll
                for (int r = 0; r < 8; ++r)
                    tri_s[(prow + r) * T_ + tcol] =
                        fmaxf(fminf(d1[r], d2[r]), 0.0f);
            }
        }
    }
    __syncthreads();

    // ---------------- Phase 3: dim-masked top-2 over pairs ----------------
    if (tid < 64) {
        const int d = tid >> 5;       // homology dim 0/1
        const int t = tid & 31;
        float top1 = 0.0f, top2 = 0.0f;   // tri >= 0 always, so 0-init is exact
#pragma unroll 4
        for (int p = 0; p < P_; ++p) {
            float v = tri_s[p * T_ + t];          // stride-1 across lanes
            v = (dim_s[p] == d) ? v : 0.0f;       // broadcast read
            if (v > top1)      { top2 = top1; top1 = v; }
            else if (v > top2) { top2 = v; }
        }
        // out[b][c][d][k][t] -> flat: bc*128 + d*64 + k*32 + t
        float* o = out + (size_t)bc * 128 + d * 64 + t;
        o[0]  = top1;   // k = 0
        o[32] = top2;   // k = 1
    }
}

extern "C" void kernel_launch(void* const* d_in, const int* in_sizes, int n_in,
                              void* d_out, int out_size, void* d_ws, size_t ws_size,
                              hipStream_t stream)
{
    const float* x         = (const float*)d_in[0];
    const int*   birth_idx = (const int*)d_in[1];
    const int*   death_idx = (const int*)d_in[2];
    const int*   pair_dim  = (const int*)d_in[3];
    float*       out       = (float*)d_out;

    dim3 grid(BC_TOTAL);   // one workgroup per (b,c)
    dim3 block(128);       // 4 wave32 waves
    CubPL2d_kernel<<<grid, block, 0, stream>>>(x, birth_idx, death_idx, pair_dim, out);
}